// SparseAttentionMeansim_70703751627039
// MI455X (gfx1250) — compile-verified
//
#include <hip/hip_runtime.h>
#include <math.h>
#include <stdint.h>

typedef __attribute__((ext_vector_type(2))) float v2f;
typedef __attribute__((ext_vector_type(8))) float v8f;

static constexpr int HN = 16;    // heads
static constexpr int LN = 2048;  // seq len
static constexpr int DN = 64;    // head dim
static constexpr int NQ = 16;    // q blocks (BQ=128)
static constexpr int NK = 32;    // k blocks (BK=64)
static constexpr float SCALE = 0.125f;  // 1/sqrt(64)

// ---------------- kernel 1: column means of K over L, per head ----------------
__global__ __launch_bounds__(64) void kmeanK(const float* __restrict__ k,
                                             float* __restrict__ kmean) {
    int h = blockIdx.x, d = threadIdx.x;
    const float* p = k + (size_t)h * LN * DN + d;
    float s = 0.f;
    for (int l = 0; l < LN; ++l) s += p[(size_t)l * DN];
    kmean[h * DN + d] = s * (1.0f / (float)LN);
}

// ------- kernel 2: center K -> kc, per k-block mean km, comp_k flag -----------
__global__ __launch_bounds__(64) void kstats(const float* __restrict__ k,
                                             const float* __restrict__ kmean,
                                             const float* __restrict__ sth,
                                             float* __restrict__ kc,
                                             float* __restrict__ km,
                                             int* __restrict__ compk) {
    int h = blockIdx.x >> 5, kb = blockIdx.x & 31;
    int d = threadIdx.x;  // 64 threads: thread = column, then thread = row
    __shared__ float lT[64][65];
    __shared__ float lM[64];
    __shared__ float lred[64];
    __shared__ float snorm;

    float mu = kmean[h * DN + d];
    const float* kp = k + ((size_t)(h * LN + kb * 64)) * DN;
    float* kcp = kc + ((size_t)(h * LN + kb * 64)) * DN;
    float cs = 0.f;
    for (int j = 0; j < 64; ++j) {
        float x = kp[(size_t)j * DN + d] - mu;
        kcp[(size_t)j * DN + d] = x;
        lT[j][d] = x;
        cs += x;
    }
    float kmd = cs * (1.0f / 64.0f);
    km[((size_t)(h * NK + kb)) * DN + d] = kmd;
    lM[d] = kmd;
    lred[d] = kmd * kmd;
    __syncthreads();
    if (d == 0) {
        float s = 0.f;
        for (int i = 0; i < 64; ++i) s += lred[i];
        snorm = sqrtf(s);
    }
    __syncthreads();
    float dot = 0.f, nrm = 0.f;
    for (int dd = 0; dd < 64; ++dd) {
        float x = lT[d][dd];
        dot += x * lM[dd];
        nrm += x * x;
    }
    float sim = dot / ((sqrtf(nrm) + 1e-6f) * (snorm + 1e-6f));
    __syncthreads();
    lred[d] = sim;
    __syncthreads();
    if (d == 0) {
        float s = 0.f;
        for (int i = 0; i < 64; ++i) s += lred[i];
        compk[h * NK + kb] = (s * (1.0f / 64.0f)) > sth[h] ? 1 : 0;
    }
}

// ------------- kernel 3: per q-block mean qm, comp_q flag ---------------------
__global__ __launch_bounds__(64) void qstats(const float* __restrict__ q,
                                             const float* __restrict__ sth,
                                             float* __restrict__ qm,
                                             int* __restrict__ compq) {
    int h = blockIdx.x >> 4, qb = blockIdx.x & 15;
    int d = threadIdx.x;  // 64
    __shared__ float lT[128][65];
    __shared__ float lM[64];
    __shared__ float lred[64];
    __shared__ float snorm;

    const float* qp = q + ((size_t)(h * LN + qb * 128)) * DN;
    float cs = 0.f;
    for (int j = 0; j < 128; ++j) {
        float x = qp[(size_t)j * DN + d];
        lT[j][d] = x;
        cs += x;
    }
    float qmd = cs * (1.0f / 128.0f);
    qm[((size_t)(h * NQ + qb)) * DN + d] = qmd;
    lM[d] = qmd;
    lred[d] = qmd * qmd;
    __syncthreads();
    if (d == 0) {
        float s = 0.f;
        for (int i = 0; i < 64; ++i) s += lred[i];
        snorm = sqrtf(s);
    }
    __syncthreads();
    float acc = 0.f;
    for (int rr = 0; rr < 2; ++rr) {
        int j = d + 64 * rr;
        float dot = 0.f, nrm = 0.f;
        for (int dd = 0; dd < 64; ++dd) {
            float x = lT[j][dd];
            dot += x * lM[dd];
            nrm += x * x;
        }
        acc += dot / ((sqrtf(nrm) + 1e-6f) * (snorm + 1e-6f));
    }
    __syncthreads();
    lred[d] = acc;
    __syncthreads();
    if (d == 0) {
        float s = 0.f;
        for (int i = 0; i < 64; ++i) s += lred[i];
        compq[h * NQ + qb] = (s * (1.0f / 128.0f)) > sth[h] ? 1 : 0;
    }
}

// ----- kernel 4: block-softmax + CDF top-mass selection; lane == k-block ------
__global__ __launch_bounds__(32) void maskk(const float* __restrict__ qm,
                                            const float* __restrict__ km,
                                            const int* __restrict__ compq,
                                            const int* __restrict__ compk,
                                            const float* __restrict__ cdf,
                                            int* __restrict__ maskg) {
    int h = blockIdx.x >> 4, qb = blockIdx.x & 15;
    int lane = threadIdx.x;  // 32 lanes == NK
    const float* qmp = qm + ((size_t)(h * NQ + qb)) * DN;
    const float* kmp = km + ((size_t)(h * NK + lane)) * DN;
    float s = 0.f;
    for (int d = 0; d < DN; ++d) s += qmp[d] * kmp[d];
    s *= SCALE;
    float mx = s;
    for (int off = 16; off >= 1; off >>= 1) mx = fmaxf(mx, __shfl_xor(mx, off, 32));
    float e = expf(s - mx);
    float sum = e;
    for (int off = 16; off >= 1; off >>= 1) sum += __shfl_xor(sum, off, 32);
    float p = e / sum;
    // mass of elements strictly before me in stable descending order
    float prefix = 0.f;
    for (int j = 0; j < 32; ++j) {
        float pj = __shfl(p, j, 32);
        if ((pj > p) || (pj == p && j < lane)) prefix += pj;
    }
    int keep = (prefix < cdf[h]) || (prefix == 0.0f);  // rank-0 always kept
    int sel = keep || !(compq[h * NQ + qb] && compk[h * NK + lane]);
    maskg[(h * NQ + qb) * NK + lane] = sel ? 1 : 0;
}

// --------- kernel 5: block-sparse flash attention with f32 WMMA ---------------
// 1 workgroup = (h, q-block): 256 threads = 8 waves; wave w owns q rows
// [qb*128 + w*16, +16). K/V tiles double-buffered in LDS and filled with
// GLOBAL_LOAD_ASYNC_TO_LDS_B128 so the next selected block streams in while
// WMMAs run on the current one (ASYNCcnt pipeline).
__global__ __launch_bounds__(256) void attn(const float* __restrict__ q,
                                            const float* __restrict__ kc,
                                            const float* __restrict__ v,
                                            const int* __restrict__ maskg,
                                            float* __restrict__ out) {
    const int h = blockIdx.x >> 4;
    const int qb = blockIdx.x & 15;
    const int tid = threadIdx.x;
    const int w = tid >> 5;
    const int lane = tid & 31;
    const int lm = lane & 15;
    const int half = lane >> 4;

    __shared__ __align__(16) float lK[2][64][68];  // centered-K tiles (dbl buf)
    __shared__ __align__(16) float lV[2][64][68];  // V tiles (dbl buf)
    __shared__ __align__(16) float lP[128][68];    // probs C->A layout relay
    __shared__ int slist[NK];
    __shared__ int scount;

    // ---- compacted list of selected k-blocks (uniform across block) ----
    if (tid == 0) {
        int c = 0;
        for (int kb = 0; kb < NK; ++kb)
            if (maskg[(h * NQ + qb) * NK + kb]) slist[c++] = kb;
        scount = c;  // >= 1 by construction of the mask
    }

    const float* kcH = kc + (size_t)h * LN * DN;  // head base (SGPR pair)
    const float* vH = v + (size_t)h * LN * DN;

    // issue one k-block's K+V tiles as 8 async b128 loads per wave
    auto issue = [&](int buf, int kb) {
        #pragma unroll
        for (int i = 0; i < 4; ++i) {
            int idx = tid + 256 * i;
            int row = idx >> 4;
            int c4 = (idx & 15) << 2;
            unsigned voff = (unsigned)(((kb * 64 + row) * DN + c4) * 4);
            unsigned ldk = (unsigned)(uintptr_t)&lK[buf][row][c4];  // LDS byte addr
            unsigned ldv = (unsigned)(uintptr_t)&lV[buf][row][c4];
            asm volatile("global_load_async_to_lds_b128 %0, %1, %2"
                         :: "v"(ldk), "v"(voff), "s"(kcH) : "memory");
            asm volatile("global_load_async_to_lds_b128 %0, %1, %2"
                         :: "v"(ldv), "v"(voff), "s"(vH) : "memory");
        }
    };

    // Q strip as 16 A-fragments (K-chunks of 4): lane holds row lm,
    // VGPR pair = d = 4c + {0,1} (lanes 0-15) or 4c + {2,3} (lanes 16-31).
    v2f aq[16];
    {
        const float* qrow = q + ((size_t)(h * LN + qb * 128 + w * 16 + lm)) * DN;
        #pragma unroll
        for (int c = 0; c < 16; ++c)
            aq[c] = *(const v2f*)(qrow + 4 * c + 2 * half);
    }

    v8f o[4];
    float m_r[8], l_r[8];
    #pragma unroll
    for (int t = 0; t < 4; ++t)
        #pragma unroll
        for (int r = 0; r < 8; ++r) o[t][r] = 0.f;
    #pragma unroll
    for (int r = 0; r < 8; ++r) { m_r[r] = -3.0e38f; l_r[r] = 0.f; }

    __syncthreads();  // slist/scount visible
    const int nsel = scount;

    issue(0, slist[0]);  // prologue prefetch

    for (int i = 0; i < nsel; ++i) {
        const int buf = i & 1;
        const bool more = (i + 1 < nsel);
        // prefetch next block into the other buffer (clobber-safe: readers of
        // that buffer finished at iteration i-1's trailing barrier)
        if (more) {
            issue(buf ^ 1, slist[i + 1]);
            asm volatile("s_wait_asynccnt 0x8" ::: "memory");  // allow prefetch in flight
        } else {
            asm volatile("s_wait_asynccnt 0x0" ::: "memory");
        }
        __syncthreads();  // current buffer complete in all waves

        // ---- S strip = Q(16xD) * Kc^T : 4 N-tiles x 16 K-chunks ----
        v8f s[4];
        #pragma unroll
        for (int t = 0; t < 4; ++t) {
            #pragma unroll
            for (int r = 0; r < 8; ++r) s[t][r] = 0.f;
            #pragma unroll
            for (int c = 0; c < 16; ++c) {
                // B frag: B[k][n] = Kc[j0+n][4c+k]
                v2f b = *(const v2f*)(&lK[buf][t * 16 + lm][4 * c + 2 * half]);
                s[t] = __builtin_amdgcn_wmma_f32_16x16x4_f32(
                    false, aq[c], false, b, (short)0, s[t], false, false);
            }
        }

        // ---- online softmax per row (rows split by lane-half) ----
        #pragma unroll
        for (int t = 0; t < 4; ++t)
            #pragma unroll
            for (int r = 0; r < 8; ++r) s[t][r] *= SCALE;

        #pragma unroll
        for (int r = 0; r < 8; ++r) {
            float mx = fmaxf(fmaxf(s[0][r], s[1][r]), fmaxf(s[2][r], s[3][r]));
            #pragma unroll
            for (int off = 1; off <= 8; off <<= 1)
                mx = fmaxf(mx, __shfl_xor(mx, off, 32));  // halves independent
            float mnew = fmaxf(m_r[r], mx);
            float alpha = __expf(m_r[r] - mnew);
            m_r[r] = mnew;
            float rs = 0.f;
            #pragma unroll
            for (int t = 0; t < 4; ++t) {
                float p = __expf(s[t][r] - mnew);
                s[t][r] = p;
                rs += p;
            }
            #pragma unroll
            for (int off = 1; off <= 8; off <<= 1) rs += __shfl_xor(rs, off, 32);
            l_r[r] = l_r[r] * alpha + rs;
            #pragma unroll
            for (int t = 0; t < 4; ++t) {
                o[t][r] *= alpha;
                lP[w * 16 + r + 8 * half][16 * t + lm] = s[t][r];  // own rows only
            }
        }

        // ---- O += P(16x64) * V(64xD); A-frags of P from LDS (same wave) ----
        #pragma unroll
        for (int c = 0; c < 16; ++c) {
            v2f a = *(const v2f*)(&lP[w * 16 + lm][4 * c + 2 * half]);
            #pragma unroll
            for (int t = 0; t < 4; ++t) {
                v2f b;
                b.x = lV[buf][4 * c + 2 * half + 0][16 * t + lm];
                b.y = lV[buf][4 * c + 2 * half + 1][16 * t + lm];
                o[t] = __builtin_amdgcn_wmma_f32_16x16x4_f32(
                    false, a, false, b, (short)0, o[t], false, false);
            }
        }

        __syncthreads();  // all reads of buf done; i+1 may prefetch over buf^1's peer
    }

    // ---- normalize + write ----
    #pragma unroll
    for (int r = 0; r < 8; ++r) {
        float inv = 1.0f / l_r[r];  // >= 1 block selected -> l > 0
        int row = qb * 128 + w * 16 + r + 8 * half;
        float* op = out + ((size_t)(h * LN + row)) * DN;
        #pragma unroll
        for (int t = 0; t < 4; ++t) op[16 * t + lm] = o[t][r] * inv;
    }
}

extern "C" void kernel_launch(void* const* d_in, const int* in_sizes, int n_in,
                              void* d_out, int out_size, void* d_ws, size_t ws_size,
                              hipStream_t stream) {
    const float* q = (const float*)d_in[0];
    const float* k = (const float*)d_in[1];
    const float* v = (const float*)d_in[2];
    const float* sth = (const float*)d_in[3];  // simthreshd1[H]
    const float* cdf = (const float*)d_in[4];  // cdfthreshd[H]
    float* out = (float*)d_out;

    // workspace layout (floats then ints), ~8.6 MB total
    float* kc = (float*)d_ws;                    // H*L*D
    float* kmean = kc + (size_t)HN * LN * DN;    // H*D
    float* qm = kmean + HN * DN;                 // H*NQ*D
    float* km = qm + HN * NQ * DN;               // H*NK*D
    int* compq = (int*)(km + HN * NK * DN);      // H*NQ
    int* compk = compq + HN * NQ;                // H*NK
    int* maskg = compk + HN * NK;                // H*NQ*NK

    kmeanK<<<HN, 64, 0, stream>>>(k, kmean);
    kstats<<<HN * NK, 64, 0, stream>>>(k, kmean, sth, kc, km, compk);
    qstats<<<HN * NQ, 64, 0, stream>>>(q, sth, qm, compq);
    maskk<<<HN * NQ, 32, 0, stream>>>(qm, km, compq, compk, cdf, maskg);
    attn<<<HN * NQ, 256, 0, stream>>>(q, kc, v, maskg, out);
}